// BehaviorLemming_30296699306445
// MI455X (gfx1250) — compile-verified
//
#include <hip/hip_runtime.h>

// world (B=16, C=20, H=512, W=512) fp32; ELEM_ID = 3.0.
// Step 1: fall_left (forward = w-1). Step 2: fall_right (forward = w+1).
// Circular rolls; H, W are powers of two so wrap = mask.

#define ELEM_ID_F 3.0f
constexpr int Bn = 16;
constexpr int Cn = 20;
constexpr int Hn = 512;
constexpr int Wn = 512;
constexpr int HW = Hn * Wn;       // 262144
constexpr int CHW = Cn * HW;

constexpr int TH = 16;                       // output rows per block (fused pass)
constexpr int TILE_ROWS = TH + 4;            // staged rows: h0-2 .. h0+TH+1
constexpr int ROW_BYTES = Wn * 4;            // 2048
constexpr int TILE_BYTES = TILE_ROWS * ROW_BYTES;   // 40960 (40 KB LDS)

__device__ __forceinline__ int wrapH(int h) { return h & (Hn - 1); }
__device__ __forceinline__ int wrapW(int w) { return w & (Wn - 1); }

// Exact blend: a, bm in {0.0, 1.0}; matches reference float arithmetic exactly.
__device__ __forceinline__ float blend1(float a, float bm, float va, float v, float vb)
{
    return (1.0f - a) * (1.0f - bm) * v + a * va + bm * vb;
}

// -------------------- kernel 1: step-1 mask (fall_left, forward = w-1) -----
__global__ void __launch_bounds__(256)
mask1_kernel(const float* __restrict__ world, unsigned char* __restrict__ a1)
{
    int t = blockIdx.x * blockDim.x + threadIdx.x;    // 0 .. H*W-1
    int b = blockIdx.y;
    int w = t & (Wn - 1);
    int h = t >> 9;
    const float* ch0 = world + (size_t)b * CHW;        // channel 0 (element id)
    const float* ch1 = ch0 + HW;                       // channel 1 (density)
    int hm1 = wrapH(h - 1), wm1 = wrapW(w - 1);
    float d   = ch1[h   * Wn + w];
    float df  = ch1[h   * Wn + wm1];
    float da  = ch1[hm1 * Wn + w];
    float dfa = ch1[hm1 * Wn + wm1];
    bool is_el = (ch0[h * Wn + w] == ELEM_ID_F);
    bool m = is_el && (df - d >= 0.0f) && (da - d < 0.0f) && (dfa - d < 0.0f);
    a1[(size_t)b * HW + t] = m ? (unsigned char)1 : (unsigned char)0;
}

// Step-1 output value of channel `chan` at (h, w), reconstructed from world + a1.
__device__ __forceinline__ float w1_at(const float* __restrict__ chan,
                                       const unsigned char* __restrict__ a1b,
                                       int h, int w)
{
    int hm1 = wrapH(h - 1), hp1 = wrapH(h + 1);
    float a  = (float)a1b[h   * Wn + w];
    float bm = (float)a1b[hp1 * Wn + w];
    return blend1(a, bm, chan[hm1 * Wn + w], chan[h * Wn + w], chan[hp1 * Wn + w]);
}

// -------------------- kernel 2: step-2 mask (fall_right, forward = w+1) ----
__global__ void __launch_bounds__(256)
mask2_kernel(const float* __restrict__ world, const unsigned char* __restrict__ a1,
             unsigned char* __restrict__ a2)
{
    int t = blockIdx.x * blockDim.x + threadIdx.x;
    int b = blockIdx.y;
    int w = t & (Wn - 1);
    int h = t >> 9;
    const float* ch0 = world + (size_t)b * CHW;
    const float* ch1 = ch0 + HW;
    const unsigned char* a1b = a1 + (size_t)b * HW;
    int hm1 = wrapH(h - 1), wp1 = wrapW(w + 1);
    float d   = w1_at(ch1, a1b, h,   w);
    float df  = w1_at(ch1, a1b, h,   wp1);
    float da  = w1_at(ch1, a1b, hm1, w);
    float dfa = w1_at(ch1, a1b, hm1, wp1);
    float el  = w1_at(ch0, a1b, h,   w);
    bool m = (el == ELEM_ID_F) && (df - d >= 0.0f) && (da - d < 0.0f) && (dfa - d < 0.0f);
    a2[(size_t)b * HW + t] = m ? (unsigned char)1 : (unsigned char)0;
}

// -------------------- kernel 3: fused step1+step2 value pass ---------------
// Each block: TH rows x full W of one (b,c) plane. World rows h0-2..h0+TH+1
// are staged into LDS with gfx1250 async global->LDS b128 copies (ASYNCcnt),
// then each output is computed from 5 LDS row-vectors (ds_load_b128).
__global__ void __launch_bounds__(256)
fused_final_kernel(const float* __restrict__ world,
                   const unsigned char* __restrict__ a1,
                   const unsigned char* __restrict__ a2,
                   float* __restrict__ out)
{
    __shared__ __align__(16) float smem[TILE_ROWS * Wn];   // 40 KB

    int tid = threadIdx.x;
    int h0  = blockIdx.x * TH;
    int c   = blockIdx.y;
    int b   = blockIdx.z;

    const float* plane  = world + (size_t)(b * Cn + c) * HW;
    float*       oplane = out   + (size_t)(b * Cn + c) * HW;
    const unsigned char* a1b = a1 + (size_t)b * HW;
    const unsigned char* a2b = a2 + (size_t)b * HW;

    // ---- async stage: 20 rows (wrapped in h) -> LDS, 16 B per lane per round
    unsigned lds_base = (unsigned)(size_t)&smem[0];   // low 32 bits = LDS offset
    unsigned long long gbase = (unsigned long long)(const void*)plane;
#pragma unroll
    for (int rnd = 0; rnd < TILE_BYTES / 4096; ++rnd) {   // 10 rounds
        int o    = rnd * 4096 + tid * 16;                 // 0 .. TILE_BYTES-16
        int r    = o >> 11;                               // staged row 0..19
        int colb = o & (ROW_BYTES - 1);
        int gh   = wrapH(h0 - 2 + r);
        unsigned goff  = (unsigned)(gh * ROW_BYTES + colb);
        unsigned laddr = lds_base + (unsigned)o;
        asm volatile("global_load_async_to_lds_b128 %0, %1, %2"
                     :
                     : "v"(laddr), "v"(goff), "s"(gbase)
                     : "memory");
    }
    asm volatile("s_wait_asynccnt 0x0" ::: "memory");   // this wave's DMAs done
    __syncthreads();                                    // all waves' DMAs done

    // Speculative prefetch of part of the next tile (global_prefetch_b8).
    if (tid < 64)
        __builtin_prefetch(plane + (size_t)wrapH(h0 + TH + 2) * Wn + tid * 8, 0, 1);

    // ---- compute: 16 rows x 128 float4 = 2048 float4 per block, 8 per thread
#pragma unroll
    for (int it = 0; it < (TH * (Wn / 4)) / 256; ++it) {  // 8 iterations
        int idx = it * 256 + tid;       // 0..2047
        int w4  = idx & (Wn / 4 - 1);   // 0..127
        int hr  = idx >> 7;             // 0..15
        int h   = h0 + hr;
        int w   = w4 * 4;

        // staged row hr+k corresponds to world row h-2+k
        const float4* col = (const float4*)&smem[(size_t)hr * Wn + w];
        float4 vm2 = col[0 * (Wn / 4)];
        float4 vm1 = col[1 * (Wn / 4)];
        float4 v0  = col[2 * (Wn / 4)];
        float4 vp1 = col[3 * (Wn / 4)];
        float4 vp2 = col[4 * (Wn / 4)];

        int hm1 = wrapH(h - 1), hp1 = wrapH(h + 1), hp2 = wrapH(h + 2);
        uchar4 a1m1 = *(const uchar4*)(a1b + hm1 * Wn + w);
        uchar4 a10  = *(const uchar4*)(a1b + h   * Wn + w);
        uchar4 a1p1 = *(const uchar4*)(a1b + hp1 * Wn + w);
        uchar4 a1p2 = *(const uchar4*)(a1b + hp2 * Wn + w);
        uchar4 a20  = *(const uchar4*)(a2b + h   * Wn + w);
        uchar4 a2p1 = *(const uchar4*)(a2b + hp1 * Wn + w);

        float Vm2[4] = {vm2.x, vm2.y, vm2.z, vm2.w};
        float Vm1[4] = {vm1.x, vm1.y, vm1.z, vm1.w};
        float V0[4]  = {v0.x,  v0.y,  v0.z,  v0.w};
        float Vp1[4] = {vp1.x, vp1.y, vp1.z, vp1.w};
        float Vp2[4] = {vp2.x, vp2.y, vp2.z, vp2.w};
        unsigned char M1[4] = {a1m1.x, a1m1.y, a1m1.z, a1m1.w};
        unsigned char M2[4] = {a10.x,  a10.y,  a10.z,  a10.w};
        unsigned char M3[4] = {a1p1.x, a1p1.y, a1p1.z, a1p1.w};
        unsigned char M4[4] = {a1p2.x, a1p2.y, a1p2.z, a1p2.w};
        unsigned char N1[4] = {a20.x,  a20.y,  a20.z,  a20.w};
        unsigned char N2[4] = {a2p1.x, a2p1.y, a2p1.z, a2p1.w};

        float r[4];
#pragma unroll
        for (int i = 0; i < 4; ++i) {
            float w1m1 = blend1((float)M1[i], (float)M2[i], Vm2[i], Vm1[i], V0[i]);
            float w10  = blend1((float)M2[i], (float)M3[i], Vm1[i], V0[i],  Vp1[i]);
            float w1p1 = blend1((float)M3[i], (float)M4[i], V0[i],  Vp1[i], Vp2[i]);
            r[i] = blend1((float)N1[i], (float)N2[i], w1m1, w10, w1p1);
        }
        *(float4*)(oplane + (size_t)h * Wn + w) = make_float4(r[0], r[1], r[2], r[3]);
    }
}

extern "C" void kernel_launch(void* const* d_in, const int* in_sizes, int n_in,
                              void* d_out, int out_size, void* d_ws, size_t ws_size,
                              hipStream_t stream)
{
    (void)in_sizes; (void)n_in; (void)out_size; (void)ws_size;
    const float* world = (const float*)d_in[0];   // only input the reference uses
    float* out = (float*)d_out;

    // Workspace: two uint8 masks of shape (B, H, W) = 4 MiB each.
    unsigned char* a1 = (unsigned char*)d_ws;
    unsigned char* a2 = a1 + (size_t)Bn * HW;

    dim3 blk(256);
    mask1_kernel<<<dim3(HW / 256, Bn), blk, 0, stream>>>(world, a1);
    mask2_kernel<<<dim3(HW / 256, Bn), blk, 0, stream>>>(world, a1, a2);
    fused_final_kernel<<<dim3(Hn / TH, Cn, Bn), blk, 0, stream>>>(world, a1, a2, out);
}